// LFQ_46110768890167
// MI455X (gfx1250) — compile-verified
//
#include <hip/hip_runtime.h>
#include <hip/hip_bf16.h>

// LFQ forward for MI455X (gfx1250, wave32).
//   z   = x @ W_in^T + b_in          [N,16]   (fp32 WMMA 16x16x4, K=2048)
//   q   = sign(z)
//   out = q @ W_out^T                [N,2048] (fp32 WMMA 16x16x4, K=16)
//   idx = sum_z (z>0) * 2^z          (wave ballot)
//
// One wave processes one 16-token tile. Memory-bound kernel (~537 MB total
// traffic -> ~23 us floor at 23.3 TB/s); weights (256 KB) stay L2-resident.

#define LFQ_D 2048
#define LFQ_Z 16

typedef float v2f __attribute__((ext_vector_type(2)));
typedef float v8f __attribute__((ext_vector_type(8)));

__global__ __launch_bounds__(256) void LFQ_46110768890167_kernel(
    const float* __restrict__ x,      // [N, 2048]
    const float* __restrict__ W_in,   // [16, 2048]
    const float* __restrict__ b_in,   // [16]
    const float* __restrict__ W_out,  // [2048, 16]
    float* __restrict__ out,          // [N, 2048]
    unsigned long long* __restrict__ idx,  // [N] (int64)
    float* __restrict__ loss,         // [1]
    int n_tiles)                      // N / 16
{
    // 8 waves per block; zbuf transposes each wave's 16x16 z tile.
    __shared__ float zbuf[8][16][16];

    const unsigned tid  = threadIdx.x;
    const unsigned lane = tid & 31u;
    const unsigned wave = tid >> 5;
    const unsigned half = lane >> 4;   // 0: K=0,1 / M rows 0-7 ; 1: K=2,3 / M rows 8-15
    const unsigned l16  = lane & 15u;

    const int tile = (int)(blockIdx.x * 8u + wave);
    if (tile >= n_tiles) return;

    // ---------------- proj_in: z = x @ W_in^T ----------------
    // A-frag (16x4 fp32): lane(l16,half) holds x[tile*16+l16][k0 + 2*half + {0,1}]
    // B-frag (4x16 fp32): lane(l16,half) holds W_in[l16][k0 + 2*half + {0,1}]
    const float* xrow = x + (size_t)(tile * 16 + (int)l16) * LFQ_D + 2u * half;
    const float* wrow = W_in + (size_t)l16 * LFQ_D + 2u * half;

    v8f c = {0.f, 0.f, 0.f, 0.f, 0.f, 0.f, 0.f, 0.f};

    for (int k0 = 0; k0 < LFQ_D; k0 += 32) {
        // Stream-ahead prefetch of the x rows, 8 cachelines ahead per lane.
        // locality=3 -> WGP-scope prefetch: pull into all cache levels (each
        // 128B line is fully consumed by the wave within one k0 iteration).
        __builtin_prefetch(xrow + k0 + 256, 0, 3);
#pragma unroll
        for (int kk = 0; kk < 32; kk += 4) {
            v2f a = *(const v2f*)(xrow + k0 + kk);
            v2f b = *(const v2f*)(wrow + k0 + kk);
            c = __builtin_amdgcn_wmma_f32_16x16x4_f32(
                    false, a, false, b, (short)0, c, false, false);
        }
    }

    // bias: C-frag column (z index) == l16 for every accumulator vgpr
    {
        float bin = b_in[l16];
#pragma unroll
        for (int i = 0; i < 8; ++i) c[i] += bin;
    }

    // ---------------- indices via wave ballot ----------------
    // C vgpr i: lanes 0-15 -> token tile*16+i (z = lane), lanes 16-31 -> token +8.
#pragma unroll
    for (int i = 0; i < 8; ++i) {
        unsigned m = (unsigned)__ballot(c[i] > 0.0f);
        if (lane == 0) {
            idx[(size_t)tile * 16 + i]     = (unsigned long long)(m & 0xFFFFu);
            idx[(size_t)tile * 16 + i + 8] = (unsigned long long)(m >> 16);
        }
    }

    // ---------------- transpose z through LDS, quantize ----------------
    // write: zbuf[w][token][z]
#pragma unroll
    for (int i = 0; i < 8; ++i)
        zbuf[wave][i + 8 * half][l16] = c[i];

    // Cross-lane LDS RAW within the same wave: wait for our DS stores to land
    // before re-reading the transposed tile (CDNA5 split counter: DScnt).
    asm volatile("s_wait_dscnt 0x0" ::: "memory");

    // A-frags of q (16x16 split into 4 K-slices of 4): lane(l16,half) needs
    // q[l16][4k + 2*half + {0,1}]
    v2f aq[4];
#pragma unroll
    for (int k = 0; k < 4; ++k) {
        float z0 = zbuf[wave][l16][4 * k + 2 * half + 0];
        float z1 = zbuf[wave][l16][4 * k + 2 * half + 1];
        aq[k].x = (z0 > 0.0f) ? 1.0f : -1.0f;
        aq[k].y = (z1 > 0.0f) ? 1.0f : -1.0f;
    }

    // ---------------- out_proj: out = q @ W_out^T ----------------
    // For each 16-wide column chunk d0: 4 accumulated WMMAs over K=16.
    // B-frag: lane(l16,half) holds W_out[(d0+l16)][4k + 2*half + {0,1}]
    const float* wo = W_out + (size_t)l16 * LFQ_Z + 2u * half;
    float* orow = out + (size_t)(tile * 16) * LFQ_D + l16;

    for (int d0 = 0; d0 < LFQ_D; d0 += 16) {
        v8f acc = {0.f, 0.f, 0.f, 0.f, 0.f, 0.f, 0.f, 0.f};
        const float* wch = wo + (size_t)d0 * LFQ_Z;
#pragma unroll
        for (int k = 0; k < 4; ++k) {
            v2f b = *(const v2f*)(wch + 4 * k);
            acc = __builtin_amdgcn_wmma_f32_16x16x4_f32(
                    false, aq[k], false, b, (short)0, acc, false, false);
        }
        // D-frag: vgpr i -> token row (i + 8*half), column d0 + l16.
        // Streaming (write-once) output -> non-temporal stores.
#pragma unroll
        for (int i = 0; i < 8; ++i) {
            __builtin_nontemporal_store(
                acc[i], orow + (size_t)(i + 8 * half) * LFQ_D + d0);
        }
    }

    if (tile == 0 && tid == 0) loss[0] = 0.0f;
}

extern "C" void kernel_launch(void* const* d_in, const int* in_sizes, int n_in,
                              void* d_out, int out_size, void* d_ws, size_t ws_size,
                              hipStream_t stream) {
    (void)n_in; (void)out_size; (void)d_ws; (void)ws_size;

    const float* x     = (const float*)d_in[0];
    const float* W_in  = (const float*)d_in[1];
    const float* b_in  = (const float*)d_in[2];
    const float* W_out = (const float*)d_in[3];

    const long long n_tok  = (long long)in_sizes[0] / LFQ_D;  // 32768
    const int       tiles  = (int)(n_tok / 16);               // 2048
    const int       blocks = (tiles + 7) / 8;                 // 8 waves/block

    // d_out = [ out: N*2048 f32 | indices: N int64 | loss: 1 f32 ]
    float*              out  = (float*)d_out;
    unsigned long long* idx  = (unsigned long long*)((char*)d_out +
                                (size_t)n_tok * LFQ_D * sizeof(float));
    float*              loss = (float*)((char*)d_out +
                                (size_t)n_tok * LFQ_D * sizeof(float) +
                                (size_t)n_tok * sizeof(unsigned long long));

    LFQ_46110768890167_kernel<<<blocks, 256, 0, stream>>>(
        x, W_in, b_in, W_out, out, idx, loss, tiles);
}